// L1AttnSparse_9818295239254
// MI455X (gfx1250) — compile-verified
//
#include <hip/hip_runtime.h>

// CDNA5 / gfx1250. wave32. WMMA f32 16x16x4 for the attn@V contraction,
// async global->LDS staging (GLOBAL_LOAD_ASYNC_TO_LDS_B128) for Q and V tiles.

typedef __attribute__((ext_vector_type(2))) float v2f;
typedef __attribute__((ext_vector_type(8))) float v8f;

#define NHEADS 8
#define WIDTH  64
#define DSTM   32          // dst_mxlen
#define GROUP  16          // dst tokens per workgroup (one WMMA M tile)
#define WIN    48          // padded source window: [d0-31, d0+15] -> 47 rows + 1 pad
#define KPAD   65          // Ks row stride (odd -> conflict-free per-lane row gather)
#define BPAD   49          // band row stride (odd -> conflict-free A-fragment reads)
#define BLOCK  128         // 4 waves

#if defined(__gfx1250__) && __has_builtin(__builtin_amdgcn_global_load_async_to_lds_b128)
#define HAVE_ASYNC_LDS 1
// Builtin expects: (int4 AS1*, int4 AS3*, imm offset, imm cpol)
typedef int v4i __attribute__((vector_size(4 * sizeof(int))));
typedef __attribute__((address_space(1))) v4i gv4i;   // global
typedef __attribute__((address_space(3))) v4i lv4i;   // LDS
// Global AS1 address == flat address numerically; LDS AS3 offset == low 32 bits
// of the flat shared-aperture address. Integer round-trips avoid addrspacecasts.
#define TO_GLOBAL(p) ((gv4i*)(uintptr_t)(p))
#define TO_LDS(p)    ((lv4i*)(uint32_t)(uintptr_t)(p))
#endif

__global__ __launch_bounds__(BLOCK)
void l1attn_sparse_wmma(const float* __restrict__ vten,
                        const float* __restrict__ qten,
                        const float* __restrict__ kten,
                        const int*   __restrict__ coo,
                        const int*   __restrict__ use_softmax_p,
                        float*       __restrict__ out,
                        int n_tok)
{
    __shared__ float Qs[GROUP * WIDTH];      // 4 KB   (stride 64, 16B aligned rows)
    __shared__ float Ks[WIN * KPAD];         // ~12.5 KB (padded stride 65)
    __shared__ float Vs[WIN * WIDTH];        // 12 KB  (stride 64: WMMA B reads conflict-free)
    __shared__ float band[GROUP * BPAD];     // ~3 KB
    __shared__ float rowsum[GROUP];

    const int g   = blockIdx.x;
    const int h   = blockIdx.y;
    const int d0  = g * GROUP;
    const int tid = threadIdx.x;
    const int t0  = d0 - (DSTM - 1);         // window start token (may be negative)

    // --- zero band / rowsum -------------------------------------------------
    for (int idx = tid; idx < GROUP * BPAD; idx += BLOCK) band[idx] = 0.f;
    if (tid < GROUP) rowsum[tid] = 0.f;

    // --- stage Q tile (16 x 64) ---------------------------------------------
    #pragma unroll
    for (int u = 0; u < 2; ++u) {
        int idx = tid + u * BLOCK;           // 256 float4's
        int row = idx >> 4, c4 = (idx & 15) << 2;
        const float* src = qten + ((size_t)(d0 + row) * NHEADS + h) * WIDTH + c4;
        float*       dst = Qs + row * WIDTH + c4;
#if HAVE_ASYNC_LDS
        __builtin_amdgcn_global_load_async_to_lds_b128(TO_GLOBAL(src), TO_LDS(dst), 0, 0);
#else
        *(float4*)dst = *(const float4*)src;
#endif
    }
    // --- stage K/V windows (48 rows x 64), clamped tokens -------------------
    #pragma unroll
    for (int u = 0; u < 6; ++u) {
        int idx = tid + u * BLOCK;           // 768 float4's each
        int j = idx >> 4, c4 = (idx & 15) << 2;
        int t = t0 + j;
        t = t < 0 ? 0 : (t >= n_tok ? n_tok - 1 : t);
        size_t base = ((size_t)t * NHEADS + h) * WIDTH + c4;
        // V: 16B-aligned LDS rows -> async direct-to-LDS when available
        float* vdst = Vs + j * WIDTH + c4;
#if HAVE_ASYNC_LDS
        __builtin_amdgcn_global_load_async_to_lds_b128(TO_GLOBAL(vten + base), TO_LDS(vdst), 0, 0);
#else
        *(float4*)vdst = *(const float4*)(vten + base);
#endif
        // K: padded stride (65 words) is only 4B-aligned -> stage via VGPRs
        float4 kv = *(const float4*)(kten + base);
        float* kd = Ks + j * KPAD + c4;
        kd[0] = kv.x; kd[1] = kv.y; kd[2] = kv.z; kd[3] = kv.w;
    }
#if HAVE_ASYNC_LDS
  #if __has_builtin(__builtin_amdgcn_s_wait_asynccnt)
    __builtin_amdgcn_s_wait_asynccnt(0);
  #else
    asm volatile("s_wait_asynccnt 0" ::: "memory");
  #endif
#endif
    __syncthreads();

    // --- phase A: L1 logits -> exp -> scatter-add into band -----------------
    const float scale = -0.125f;             // -1/sqrt(64)
    #pragma unroll
    for (int u = 0; u < 4; ++u) {
        int p  = tid + u * BLOCK;            // (i,c) pair, 512 total
        int rr = d0 * DSTM + p;              // coo rows for this group are contiguous
        int dst = coo[rr * 3 + 0];
        int src = coo[rr * 3 + 1];
        int i = dst - d0;
        int j = src - t0;
        j = j < 0 ? 0 : (j > WIN - 1 ? WIN - 1 : j);
        if ((unsigned)i < (unsigned)GROUP) {
            const float* qrow = Qs + i * WIDTH;
            const float* krow = Ks + j * KPAD;
            float s = 0.f;
            #pragma unroll
            for (int w = 0; w < WIDTH; ++w) s += fabsf(qrow[w] - krow[w]);
            float e = __expf(s * scale);
            atomicAdd(&rowsum[i], e);            // ds_add_f32
            atomicAdd(&band[i * BPAD + j], e);   // scatter-add handles clipped dup srcs
        }
    }
    __syncthreads();

    // --- softmax normalize (baseline slot contributes exp(0)=1) -------------
    const int usesm = *use_softmax_p;
    for (int idx = tid; idx < GROUP * WIN; idx += BLOCK) {
        int i = idx / WIN, jj = idx - i * WIN;
        float nrm = usesm ? 1.f / (1.f + rowsum[i]) : 1.f;
        band[i * BPAD + jj] *= nrm;
    }
    __syncthreads();

    // --- phase B: out(16x64) = band(16x48) @ Vwin(48x64) via WMMA f32 -------
    // one 16x16 N-tile per wave; 12 K-steps of 4.
    const int lane = tid & 31;
    const int n0   = (tid >> 5) * 16;
    const int lm   = lane & 15;
    const bool hiL = lane >= 16;

    v8f acc = {};
    #pragma unroll
    for (int k0 = 0; k0 < WIN; k0 += 4) {
        // A 16x4 layout: VGPR0 = K={0|2} per lane half, VGPR1 = K={1|3}
        int ka = k0 + (hiL ? 2 : 0);
        v2f a, b;
        a.x = band[lm * BPAD + ka];
        a.y = band[lm * BPAD + ka + 1];
        // B 4x16 layout mirrors A: rows K={0|2} in VGPR0 halves, K={1|3} in VGPR1
        b.x = Vs[ka * WIDTH + n0 + lm];
        b.y = Vs[(ka + 1) * WIDTH + n0 + lm];
        acc = __builtin_amdgcn_wmma_f32_16x16x4_f32(
            /*neg_a=*/false, a, /*neg_b=*/false, b,
            /*c_mod=*/(short)0, acc, /*reuse_a=*/false, /*reuse_b=*/false);
    }

    // C/D layout: VGPR r -> rows m=r (lanes 0-15) / m=r+8 (lanes 16-31), col = lane%16
    #pragma unroll
    for (int r = 0; r < 8; ++r) {
        int m = r + (hiL ? 8 : 0);
        out[((size_t)(d0 + m) * NHEADS + h) * WIDTH + n0 + lm] = acc[r];
    }
}

extern "C" void kernel_launch(void* const* d_in, const int* in_sizes, int n_in,
                              void* d_out, int out_size, void* d_ws, size_t ws_size,
                              hipStream_t stream) {
    // setup_inputs order: v, q, k, coo, dst_mxlen, src_mxlen, use_softmax
    const float* v   = (const float*)d_in[0];
    const float* q   = (const float*)d_in[1];
    const float* k   = (const float*)d_in[2];
    const int*   coo = (const int*)d_in[3];
    const int*   usm = (const int*)d_in[6];
    float* out = (float*)d_out;

    int n_tok = in_sizes[1] / (NHEADS * WIDTH);   // q flat size / (heads*width)
    dim3 grid(n_tok / GROUP, NHEADS);
    l1attn_sparse_wmma<<<grid, BLOCK, 0, stream>>>(v, q, k, coo, usm, out, n_tok);
}